// CayleyConv_3590592659589
// MI455X (gfx1250) — compile-verified
//
#include <hip/hip_runtime.h>
#include <hip/hip_bf16.h>
#include <stdint.h>

// CayleyConv for MI455X (gfx1250). fp32 throughout (matches reference dtype).
// Dense GEMMs use V_WMMA_F32_16X16X4_F32; SpMV uses COO scatter with f32
// global atomics (y panels fit in the 192MB L2, so gathers mostly hit L2).

typedef float v2f __attribute__((ext_vector_type(2)));
typedef float v8f __attribute__((ext_vector_type(8)));

#define C_DIM 128
#define JACOBI_ITERS 10

// ---------------- elementwise / setup kernels ----------------

__global__ void fill0_k(float* __restrict__ p, size_t n) {
  size_t i = (size_t)blockIdx.x * blockDim.x + threadIdx.x;
  if (i < n) p[i] = 0.0f;
}

__global__ void deg_scatter_k(const long long* __restrict__ ei,
                              const float* __restrict__ ew,
                              float* __restrict__ deg, int E) {
  int e = blockIdx.x * blockDim.x + threadIdx.x;
  if (e >= E) return;
  atomicAdd(&deg[ei[e]], ew[e]);
}

__global__ void dis_k(float* __restrict__ deg, int n) {
  int i = blockIdx.x * blockDim.x + threadIdx.x;
  if (i >= n) return;
  float d = deg[i];
  deg[i] = (d > 0.0f) ? rsqrtf(d) : 0.0f;   // deg^-0.5, isolated nodes -> 0
}

// wf[e] = -h * dis[row]*w*dis[col]; self-edges also accumulate into sd[row]
__global__ void edgew_k(const long long* __restrict__ ei,
                        const float* __restrict__ ew,
                        const float* __restrict__ dis,
                        const float* __restrict__ hp,
                        float* __restrict__ wf, float* __restrict__ sd, int E) {
  int e = blockIdx.x * blockDim.x + threadIdx.x;
  if (e >= E) return;
  long long r = ei[e];
  long long c = ei[(size_t)E + e];
  float w = -hp[0] * dis[r] * ew[e] * dis[c];
  wf[e] = w;
  if (r == c) atomicAdd(&sd[r], w);
}

// diag(A) = (h + sd) - 1j ; 1/d = (dre + 1j) / (dre^2 + 1)
__global__ void invd_k(const float* __restrict__ sd, const float* __restrict__ hp,
                       float* __restrict__ ir, float* __restrict__ ii, int n) {
  int i = blockIdx.x * blockDim.x + threadIdx.x;
  if (i >= n) return;
  float dre = hp[0] + sd[i];
  float den = dre * dre + 1.0f;               // |d|^2 >= 1, never singular
  ir[i] = dre / den;
  ii[i] = 1.0f / den;
}

__global__ void yinit_k(const float* __restrict__ x,
                        float* __restrict__ yr, float* __restrict__ yi, size_t nc) {
  size_t i = (size_t)blockIdx.x * blockDim.x + threadIdx.x;
  if (i >= nc) return;
  yr[i] = x[i];
  yi[i] = 0.0f;
}

// b seed: (h + 1j) * y  (SpMV scatter of S*y is added on top afterwards)
__global__ void binit_k(const float* __restrict__ yr, const float* __restrict__ yi,
                        const float* __restrict__ hp,
                        float* __restrict__ br, float* __restrict__ bi, size_t nc) {
  size_t i = (size_t)blockIdx.x * blockDim.x + threadIdx.x;
  if (i >= nc) return;
  float h = hp[0];
  float r = yr[i], m = yi[i];
  br[i] = h * r - m;
  bi[i] = h * m + r;
}

// yk0 = inv_d * b ; seed t = -(b + sd*yk0)
__global__ void jstart_k(const float* __restrict__ br, const float* __restrict__ bi,
                         const float* __restrict__ ir, const float* __restrict__ ii,
                         const float* __restrict__ sd,
                         float* __restrict__ yr, float* __restrict__ yi,
                         float* __restrict__ tr, float* __restrict__ ti, size_t nc) {
  size_t i = (size_t)blockIdx.x * blockDim.x + threadIdx.x;
  if (i >= nc) return;
  int node = (int)(i >> 7);                    // /128
  float R = ir[node], I = ii[node], S = sd[node];
  float ur = br[i], ui = bi[i];
  float y_r = R * ur - I * ui;
  float y_i = R * ui + I * ur;
  yr[i] = y_r; yi[i] = y_i;
  tr[i] = -(br[i] + S * y_r);
  ti[i] = -(bi[i] + S * y_i);
}

// after scatter, t = S*yk - b - sd*yk = -u ; yk_new = inv_d * u ; reseed t
__global__ void combine_init_k(const float* __restrict__ br, const float* __restrict__ bi,
                               const float* __restrict__ ir, const float* __restrict__ ii,
                               const float* __restrict__ sd,
                               float* __restrict__ tr, float* __restrict__ ti,
                               float* __restrict__ yr, float* __restrict__ yi, size_t nc) {
  size_t i = (size_t)blockIdx.x * blockDim.x + threadIdx.x;
  if (i >= nc) return;
  int node = (int)(i >> 7);
  float R = ir[node], I = ii[node], S = sd[node];
  float ur = -tr[i], ui = -ti[i];
  float y_r = R * ur - I * ui;
  float y_i = R * ui + I * ur;
  yr[i] = y_r; yi[i] = y_i;
  tr[i] = -(br[i] + S * y_r);
  ti[i] = -(bi[i] + S * y_i);
}

// ---------------- SpMV: COO scatter, one wave32 per edge ----------------

__global__ void __launch_bounds__(256)
spmv_scatter_k(const long long* __restrict__ ei, const float* __restrict__ wf,
               const float* __restrict__ xr, const float* __restrict__ xi,
               float* __restrict__ outr, float* __restrict__ outi, int E) {
  int e = blockIdx.x * 8 + (threadIdx.x >> 5);   // 8 edges per 256-thread block
  if (e >= E) return;
  int lane = threadIdx.x & 31;
  long long r = ei[e];
  long long c = ei[(size_t)E + e];
  float wt = wf[e];
  const float4 vr = *(const float4*)(xr + (size_t)c * C_DIM + lane * 4);
  const float4 vi = *(const float4*)(xi + (size_t)c * C_DIM + lane * 4);
  float* dr = outr + (size_t)r * C_DIM + lane * 4;
  float* di = outi + (size_t)r * C_DIM + lane * 4;
  atomicAdd(dr + 0, wt * vr.x);
  atomicAdd(dr + 1, wt * vr.y);
  atomicAdd(dr + 2, wt * vr.z);
  atomicAdd(dr + 3, wt * vr.w);
  atomicAdd(di + 0, wt * vi.x);
  atomicAdd(di + 1, wt * vi.y);
  atomicAdd(di + 2, wt * vi.z);
  atomicAdd(di + 3, wt * vi.w);
}

// ---------------- WMMA GEMMs: out = X @ W^T (M x 128 @ 128 x 128) ----------------
// One wave32 per 16x16 tile; 8 waves/block cover all 8 N-tiles of one M-tile.
// A(16x4 f32): lane l holds A[l&15][k + (l>>4)*2 .. +1] (2 VGPRs).
// B(4x16):     lane l holds B[k + (l>>4)*2 .. +1][l&15] = W[n][k..] contiguous.
// C/D:         VGPR j -> M = j + (lane>=16 ? 8 : 0), N = lane&15.

__global__ void __launch_bounds__(256)
gemm_w0_k(const float* __restrict__ X, const float* __restrict__ W,
          float* __restrict__ out) {
  int m0 = blockIdx.x * 16;
  int lane = threadIdx.x & 31;
  int n0 = (threadIdx.x >> 5) * 16;
  int lm = lane & 15;
  int kh = lane >> 4;
  const float* arow = X + (size_t)(m0 + lm) * C_DIM;
  const float* brow = W + (size_t)(n0 + lm) * C_DIM;
  v8f c = {};
  for (int k = 0; k < C_DIM; k += 4) {
    int kk = k + kh * 2;
    v2f a = *(const v2f*)(arow + kk);
    v2f b = *(const v2f*)(brow + kk);
    c = __builtin_amdgcn_wmma_f32_16x16x4_f32(false, a, false, b, (short)0, c,
                                              false, false);
  }
  float* o = out + (size_t)(m0 + kh * 8) * C_DIM + n0 + lm;
#pragma unroll
  for (int j = 0; j < 8; ++j) o[(size_t)j * C_DIM] = c[j];
}

// out += 2 * (yr @ Wre^T - yi @ Wim^T)   (only Re(cum) reaches the output)
__global__ void __launch_bounds__(256)
gemm_acc_k(const float* __restrict__ Ar, const float* __restrict__ Ai,
           const float* __restrict__ Wre, const float* __restrict__ Wim,
           float* __restrict__ out) {
  int m0 = blockIdx.x * 16;
  int lane = threadIdx.x & 31;
  int n0 = (threadIdx.x >> 5) * 16;
  int lm = lane & 15;
  int kh = lane >> 4;
  const float* ar  = Ar  + (size_t)(m0 + lm) * C_DIM;
  const float* ai  = Ai  + (size_t)(m0 + lm) * C_DIM;
  const float* wre = Wre + (size_t)(n0 + lm) * C_DIM;
  const float* wim = Wim + (size_t)(n0 + lm) * C_DIM;
  v8f c = {};
  for (int k = 0; k < C_DIM; k += 4) {
    int kk = k + kh * 2;
    v2f a0 = *(const v2f*)(ar + kk);
    v2f b0 = *(const v2f*)(wre + kk);
    c = __builtin_amdgcn_wmma_f32_16x16x4_f32(false, a0, false, b0, (short)0, c,
                                              false, false);
    v2f a1 = *(const v2f*)(ai + kk);
    v2f b1 = -(*(const v2f*)(wim + kk));   // f32 WMMA has no A/B neg -> VALU negate
    c = __builtin_amdgcn_wmma_f32_16x16x4_f32(false, a1, false, b1, (short)0, c,
                                              false, false);
  }
  float* o = out + (size_t)(m0 + kh * 8) * C_DIM + n0 + lm;
#pragma unroll
  for (int j = 0; j < 8; ++j) o[(size_t)j * C_DIM] += 2.0f * c[j];
}

// scalar tails for M % 16 != 0 (unused for N=50000 but kept for robustness)
__global__ void gemm_w0_tail_k(const float* __restrict__ X, const float* __restrict__ W,
                               float* __restrict__ out, int mstart, int n) {
  int idx = blockIdx.x * blockDim.x + threadIdx.x;
  int rows = n - mstart;
  if (idx >= rows * C_DIM) return;
  int m = mstart + idx / C_DIM, nn = idx % C_DIM;
  float acc = 0.0f;
  for (int k = 0; k < C_DIM; ++k)
    acc += X[(size_t)m * C_DIM + k] * W[(size_t)nn * C_DIM + k];
  out[(size_t)m * C_DIM + nn] = acc;
}

__global__ void gemm_acc_tail_k(const float* __restrict__ Ar, const float* __restrict__ Ai,
                                const float* __restrict__ Wre, const float* __restrict__ Wim,
                                float* __restrict__ out, int mstart, int n) {
  int idx = blockIdx.x * blockDim.x + threadIdx.x;
  int rows = n - mstart;
  if (idx >= rows * C_DIM) return;
  int m = mstart + idx / C_DIM, nn = idx % C_DIM;
  float acc = 0.0f;
  for (int k = 0; k < C_DIM; ++k)
    acc += Ar[(size_t)m * C_DIM + k] * Wre[(size_t)nn * C_DIM + k]
         - Ai[(size_t)m * C_DIM + k] * Wim[(size_t)nn * C_DIM + k];
  out[(size_t)m * C_DIM + nn] += 2.0f * acc;
}

// ---------------- driver ----------------

extern "C" void kernel_launch(void* const* d_in, const int* in_sizes, int n_in,
                              void* d_out, int out_size, void* d_ws, size_t ws_size,
                              hipStream_t stream) {
  const float*     x   = (const float*)d_in[0];
  const long long* ei  = (const long long*)d_in[1];   // int64 (2, E)
  const float*     ew  = (const float*)d_in[2];
  const float*     hp  = (const float*)d_in[3];
  const float*     W0  = (const float*)d_in[4];
  const float*     Wre = (const float*)d_in[5];
  const float*     Wim = (const float*)d_in[6];
  float*           out = (float*)d_out;

  const int    n  = in_sizes[0] / C_DIM;              // 50000
  const int    E  = in_sizes[2];                      // 1600000
  const int    R  = in_sizes[5] / (C_DIM * C_DIM);    // 3
  const size_t NC = (size_t)n * C_DIM;

  // carve workspace (all fp32)
  char* ws = (char*)d_ws;
  size_t off = 0;
  auto carve = [&](size_t elems) -> float* {
    off = (off + 255) & ~(size_t)255;
    float* p = (float*)(ws + off);
    off += elems * sizeof(float);
    return p;
  };
  float* dis = carve(n);    // deg, then deg^-0.5 in place
  float* sd  = carve(n);    // self-edge diag accumulation
  float* ir  = carve(n);    // Re(1/d)
  float* ii  = carve(n);    // Im(1/d)
  float* wf  = carve(E);    // normalized edge weights -h*w_norm
  float* br  = carve(NC);  float* bi = carve(NC);
  float* tr  = carve(NC);  float* ti = carve(NC);
  float* yr  = carve(NC);  float* yi = carve(NC);
  (void)ws_size; (void)n_in; (void)out_size;

  const dim3 B(256);
  const unsigned gE   = (unsigned)((E + 255) / 256);
  const unsigned gN   = (unsigned)((n + 255) / 256);
  const unsigned gNC  = (unsigned)((NC + 255) / 256);
  const unsigned gEw  = (unsigned)((E + 7) / 8);      // 8 edge-waves / block
  const int      m16  = n / 16;
  const int      mrem = n - m16 * 16;

  // graph setup
  fill0_k<<<gN, B, 0, stream>>>(dis, (size_t)n);
  fill0_k<<<gN, B, 0, stream>>>(sd, (size_t)n);
  deg_scatter_k<<<gE, B, 0, stream>>>(ei, ew, dis, E);
  dis_k<<<gN, B, 0, stream>>>(dis, n);
  edgew_k<<<gE, B, 0, stream>>>(ei, ew, dis, hp, wf, sd, E);
  invd_k<<<gN, B, 0, stream>>>(sd, hp, ir, ii, n);

  // y0 = x (complex, zero imag); base term out = x @ W0^T
  yinit_k<<<gNC, B, 0, stream>>>(x, yr, yi, NC);
  if (m16 > 0) gemm_w0_k<<<(unsigned)m16, B, 0, stream>>>(x, W0, out);
  if (mrem > 0) {
    unsigned g = (unsigned)((mrem * C_DIM + 255) / 256);
    gemm_w0_tail_k<<<g, B, 0, stream>>>(x, W0, out, m16 * 16, n);
  }

  for (int r = 0; r < R; ++r) {
    // b = S*y + (h + 1j)*y
    binit_k<<<gNC, B, 0, stream>>>(yr, yi, hp, br, bi, NC);
    spmv_scatter_k<<<gEw, B, 0, stream>>>(ei, wf, yr, yi, br, bi, E);
    // y = D^-1 b ; t seeded
    jstart_k<<<gNC, B, 0, stream>>>(br, bi, ir, ii, sd, yr, yi, tr, ti, NC);
    for (int j = 0; j < JACOBI_ITERS; ++j) {
      spmv_scatter_k<<<gEw, B, 0, stream>>>(ei, wf, yr, yi, tr, ti, E);
      combine_init_k<<<gNC, B, 0, stream>>>(br, bi, ir, ii, sd, tr, ti, yr, yi, NC);
    }
    // out += 2 * (yr @ Wre_r^T - yi @ Wim_r^T)
    const float* wre = Wre + (size_t)r * C_DIM * C_DIM;
    const float* wim = Wim + (size_t)r * C_DIM * C_DIM;
    if (m16 > 0) gemm_acc_k<<<(unsigned)m16, B, 0, stream>>>(yr, yi, wre, wim, out);
    if (mrem > 0) {
      unsigned g = (unsigned)((mrem * C_DIM + 255) / 256);
      gemm_acc_tail_k<<<g, B, 0, stream>>>(yr, yi, wre, wim, out, m16 * 16, n);
    }
  }
}